// Gate_52527450030688
// MI455X (gfx1250) — compile-verified
//
#include <hip/hip_runtime.h>
#include <math.h>

// Problem constants (from reference)
#define BATCH 16384
#define NEXP  12
#define DIM   768
#define TWOD  1536

typedef __attribute__((ext_vector_type(2))) float v2f;
typedef __attribute__((ext_vector_type(8))) float v8f;

#define WAVES_PER_BLOCK 4

// ---------------------------------------------------------------------------
// Kernel 1: gating. One wave32 computes a 16-row tile of both logit matrices
// with V_WMMA_F32_16X16X4_F32 (K = 1536, i.e. 2*384 wmma steps, A shared
// between the two gates), then does the gumbel-softmax straight-through
// per row and emits ret_a / ret_i plus (idx, scale) for the gather kernel.
// ---------------------------------------------------------------------------
__global__ __launch_bounds__(32 * WAVES_PER_BLOCK) void gate_kernel(
    const float* __restrict__ audio, const float* __restrict__ image,
    const float* __restrict__ Wa, const float* __restrict__ ba,
    const float* __restrict__ Wi, const float* __restrict__ bi,
    const float* __restrict__ ga, const float* __restrict__ gi,
    float* __restrict__ retA, float* __restrict__ retI,
    int* __restrict__ idxA, int* __restrict__ idxI,
    float* __restrict__ sclA, float* __restrict__ sclI)
{
    __shared__ float lds[WAVES_PER_BLOCK][2][16][16];

    const int lane  = threadIdx.x & 31;
    const int wave  = threadIdx.x >> 5;
    const int tile  = blockIdx.x * WAVES_PER_BLOCK + wave;
    const int row   = lane & 15;      // A: M index, B: N index
    const int khalf = lane >> 4;      // K sub-phase (lanes 16-31 hold K+2)
    const int b     = tile * 16 + row;

    // A-matrix rows: gate_input[b] = [audio[b,0,:] | image[b,0,:]]
    const float* arow = audio + (size_t)b * (NEXP * DIM);
    const float* irow = image + (size_t)b * (NEXP * DIM);

    // B-matrix columns: expert n = row (zero-padded for n >= 12)
    const int   n     = (row < NEXP) ? row : 0;
    const float nmask = (row < NEXP) ? 1.0f : 0.0f;
    const float* wa = Wa + n * TWOD;
    const float* wi = Wi + n * TWOD;

    v8f ca = {};
    v8f ci = {};

    // K = 0..767 from audio[:,0,:]
    for (int k0 = 0; k0 < DIM; k0 += 4) {
        const int k = k0 + 2 * khalf;                 // per ISA 16x4 layout
        v2f a  = *(const v2f*)(arow + k);
        v2f bA = *(const v2f*)(wa + k);
        v2f bI = *(const v2f*)(wi + k);
        bA.x *= nmask; bA.y *= nmask;
        bI.x *= nmask; bI.y *= nmask;
        ca = __builtin_amdgcn_wmma_f32_16x16x4_f32(false, a, false, bA,
                                                   (short)0, ca, false, false);
        ci = __builtin_amdgcn_wmma_f32_16x16x4_f32(false, a, false, bI,
                                                   (short)0, ci, false, false);
    }
    // K = 768..1535 from image[:,0,:]
    for (int k0 = 0; k0 < DIM; k0 += 4) {
        const int k = k0 + 2 * khalf;
        v2f a  = *(const v2f*)(irow + k);
        v2f bA = *(const v2f*)(wa + DIM + k);
        v2f bI = *(const v2f*)(wi + DIM + k);
        bA.x *= nmask; bA.y *= nmask;
        bI.x *= nmask; bI.y *= nmask;
        ca = __builtin_amdgcn_wmma_f32_16x16x4_f32(false, a, false, bA,
                                                   (short)0, ca, false, false);
        ci = __builtin_amdgcn_wmma_f32_16x16x4_f32(false, a, false, bI,
                                                   (short)0, ci, false, false);
    }

    // Spill logits tile to LDS. C layout: VGPR v -> M = v + 8*khalf, N = row.
#pragma unroll
    for (int v = 0; v < 8; ++v) {
        lds[wave][0][v + 8 * khalf][row] = ca[v];
        lds[wave][1][v + 8 * khalf][row] = ci[v];
    }
    __syncthreads();

    // Phase 2: lanes 0-15 -> audio gate rows 0..15, lanes 16-31 -> image gate.
    const int    rb  = tile * 16 + row;
    const float* gg  = khalf ? gi : ga;
    const float* bbv = khalf ? bi : ba;

    float z[NEXP];
    float m = -3.402823466e38f;
#pragma unroll
    for (int e = 0; e < NEXP; ++e) {
        float l = lds[wave][khalf][row][e] + bbv[e] + gg[(size_t)rb * NEXP + e];
        z[e] = l;                       // tau = 1.0 -> no division
        m = fmaxf(m, l);
    }
    float sum = 0.0f;
    int   amax = 0;
    float best = z[0];
#pragma unroll
    for (int e = 0; e < NEXP; ++e) {
        sum += expf(z[e] - m);
        if (z[e] > best) { best = z[e]; amax = e; }   // first-max, like argmax
    }
    const float ysel = 1.0f / sum;                    // exp(0)/sum at argmax
    const float rsel = (1.0f - ysel) + ysel;          // straight-through value
    // non-argmax entries: (0 - y) + y == exactly +0.0 in fp32

    float* ret = (khalf ? retI : retA) + (size_t)rb * NEXP;
#pragma unroll
    for (int e = 0; e < NEXP; ++e) ret[e] = (e == amax) ? rsel : 0.0f;

    const float scl = rsel * (1.0f / 12.0f);          // fold the mean's 1/12
    if (khalf) { idxI[rb] = amax; sclI[rb] = scl; }
    else       { idxA[rb] = amax; sclA[rb] = scl; }
}

// ---------------------------------------------------------------------------
// Kernel 2: gather the single selected expert row per (b, modality), scale by
// ret_sel/12 and stream out [B,1536]. Only non-selected terms are exact zeros
// in the reference's mean, so this reproduces it while reading 200 MB instead
// of 1.2 GB.
// ---------------------------------------------------------------------------
__global__ __launch_bounds__(256) void gather_kernel(
    const float* __restrict__ audio, const float* __restrict__ image,
    const int* __restrict__ idxA, const int* __restrict__ idxI,
    const float* __restrict__ sclA, const float* __restrict__ sclI,
    float* __restrict__ out)
{
    const unsigned t = blockIdx.x * 256u + threadIdx.x;   // float4 index
    if (t >= (unsigned)BATCH * (TWOD / 4)) return;
    const unsigned b  = t / (TWOD / 4);
    const unsigned c4 = t - b * (TWOD / 4);

    const float4* src;
    float s;
    if (c4 < (DIM / 4)) {
        const int e = idxA[b];
        s = sclA[b];
        src = (const float4*)(audio + ((size_t)b * NEXP + e) * DIM) + c4;
    } else {
        const int e = idxI[b];
        s = sclI[b];
        src = (const float4*)(image + ((size_t)b * NEXP + e) * DIM) + (c4 - DIM / 4);
    }
    const float4 v = *src;
    float4 o;
    o.x = v.x * s; o.y = v.y * s; o.z = v.z * s; o.w = v.w * s;
    ((float4*)out)[t] = o;
}

// ---------------------------------------------------------------------------
extern "C" void kernel_launch(void* const* d_in, const int* in_sizes, int n_in,
                              void* d_out, int out_size, void* d_ws, size_t ws_size,
                              hipStream_t stream) {
    const float* audio = (const float*)d_in[0];
    const float* image = (const float*)d_in[1];
    const float* Wa    = (const float*)d_in[2];
    const float* ba    = (const float*)d_in[3];
    const float* Wi    = (const float*)d_in[4];
    const float* bi    = (const float*)d_in[5];
    const float* ga    = (const float*)d_in[6];
    const float* gi    = (const float*)d_in[7];

    float* out  = (float*)d_out;
    float* retA = out + (size_t)BATCH * TWOD;          // tuple element 2
    float* retI = retA + (size_t)BATCH * NEXP;         // tuple element 3

    int*   idxA = (int*)d_ws;
    int*   idxI = idxA + BATCH;
    float* sclA = (float*)(idxI + BATCH);
    float* sclI = sclA + BATCH;

    const int tiles  = BATCH / 16;                     // 1024
    const int blocks = tiles / WAVES_PER_BLOCK;        // 256
    gate_kernel<<<blocks, 32 * WAVES_PER_BLOCK, 0, stream>>>(
        audio, image, Wa, ba, Wi, bi, ga, gi,
        retA, retI, idxA, idxI, sclA, sclI);

    const int n4 = BATCH * (TWOD / 4);                 // 6,291,456
    gather_kernel<<<n4 / 256, 256, 0, stream>>>(
        audio, image, idxA, idxI, sclA, sclI, out);
}